// PathomicGraphNet_33457795236062
// MI455X (gfx1250) — compile-verified
//
#include <hip/hip_runtime.h>

// PathomicGraphNet for MI455X (gfx1250).
// Memory-bound problem (~6 GB traffic -> ~260us @ 23.3 TB/s); GEMMs (33 GFLOP)
// done in exact fp32 via V_WMMA_F32_16X16X4_F32 (precision is free on roofline).

#define DF 128      // feature dim
#define AHF 256     // attention hidden
#define NGROUP 16   // graphs
#define LAYERS 3
#define PADR 132    // LDS row stride in floats: bank stride 4 -> conflict-free A-frag reads

typedef float v2f __attribute__((ext_vector_type(2)));
typedef float v8f __attribute__((ext_vector_type(8)));

#if __has_builtin(__builtin_amdgcn_wmma_f32_16x16x4_f32)
static __device__ __forceinline__ v8f wmma_f32(v2f a, v2f b, v8f c) {
  // 8 args: (neg_a, A, neg_b, B, c_mod, C, reuse_a, reuse_b)
  return __builtin_amdgcn_wmma_f32_16x16x4_f32(false, a, false, b, (short)0, c,
                                               false, false);
}
#else
// Fallback stub so the compile survives; histogram wmma==0 signals to switch path.
static __device__ __forceinline__ v8f wmma_f32(v2f a, v2f b, v8f c) { return c; }
#endif

// ---------------------------------------------------------------- utilities
__global__ void fill_kernel(float* __restrict__ p, float v, long n) {
  long i = (long)blockIdx.x * blockDim.x + threadIdx.x;
  if (i < n) p[i] = v;
}

__global__ void deg_kernel(const int* __restrict__ dst, float* __restrict__ deg, int E) {
  long i = (long)blockIdx.x * blockDim.x + threadIdx.x;
  if (i < E) atomicAdd(&deg[dst[i]], 1.0f);
}

__global__ void invdeg_kernel(const float* __restrict__ deg, float* __restrict__ invd, int N) {
  long i = (long)blockIdx.x * blockDim.x + threadIdx.x;
  if (i < N) invd[i] = 1.0f / fmaxf(deg[i], 1.0f);
}

// msg[dst[e], :] += h[src[e], :]   (32 lanes per edge, 4 floats per lane)
__global__ void scatter_kernel(const float* __restrict__ h, const int* __restrict__ src,
                               const int* __restrict__ dst, float* __restrict__ msg, int E) {
  long tid = (long)blockIdx.x * blockDim.x + threadIdx.x;
  long e = tid >> 5;
  if (e < E) {
    int c = (int)(tid & 31) << 2;
    long s = src[e], d = dst[e];
    const float4 v = *(const float4*)(h + s * DF + c);
    float* p = msg + d * DF + c;
    atomicAdd(p + 0, v.x);
    atomicAdd(p + 1, v.y);
    atomicAdd(p + 2, v.z);
    atomicAdd(p + 3, v.w);
  }
}

// ------------------------------------------------- fused GNN layer (WMMA fp32)
// h_out[i,:] = relu( h_in[i,:]@Ws + bs + (msg[i,:]*invd[i])@Wn + bn ), in place.
// Block = 256 threads (8 waves) = one 16-row tile; wave w owns output cols [16w,16w+16).
__global__ __launch_bounds__(256) void layer_kernel(
    const float* __restrict__ hin, const float* __restrict__ msg,
    const float* __restrict__ invd,
    const float* __restrict__ Ws, const float* __restrict__ bs,
    const float* __restrict__ Wn, const float* __restrict__ bn,
    float* __restrict__ hout, int N) {
  __shared__ float hT[16 * PADR];
  __shared__ float mT[16 * PADR];
  const int tid = threadIdx.x;
  const int wave = tid >> 5, lane = tid & 31;
  const int m = lane & 15, hf = lane >> 4;  // A/B/C half-wave split
  const long row0 = (long)blockIdx.x * 16;

  // stage 16x128 h and scaled msg (512 float4 slots over 256 threads)
  for (int idx = tid; idx < 512; idx += 256) {
    int r = idx >> 5, c4 = (idx & 31) << 2;
    long g = row0 + r;
    if (g >= N) g = N - 1;
    const float4 hv = *(const float4*)(hin + g * DF + c4);
    const float4 mv = *(const float4*)(msg + g * DF + c4);
    float sc = invd[g];
    float* hp = &hT[r * PADR + c4];
    hp[0] = hv.x; hp[1] = hv.y; hp[2] = hv.z; hp[3] = hv.w;
    float* mp = &mT[r * PADR + c4];
    mp[0] = mv.x * sc; mp[1] = mv.y * sc; mp[2] = mv.z * sc; mp[3] = mv.w * sc;
  }
  __syncthreads();

  const int col = (wave << 4) + m;
  const float bias = bs[col] + bn[col];
  v8f acc;
#pragma unroll
  for (int r = 0; r < 8; ++r) acc[r] = bias;

#pragma unroll
  for (int k0 = 0; k0 < DF; k0 += 4) {
    const int kb = k0 + (hf << 1);  // half 0 -> K{0,1}, half 1 -> K{2,3}
    v2f a, b;
    a.x = hT[m * PADR + kb];
    a.y = hT[m * PADR + kb + 1];
    b.x = Ws[kb * DF + col];
    b.y = Ws[(kb + 1) * DF + col];
    acc = wmma_f32(a, b, acc);
    a.x = mT[m * PADR + kb];
    a.y = mT[m * PADR + kb + 1];
    b.x = Wn[kb * DF + col];
    b.y = Wn[(kb + 1) * DF + col];
    acc = wmma_f32(a, b, acc);
  }

#pragma unroll
  for (int r = 0; r < 8; ++r) {
    long g = row0 + r + (hf << 3);
    if (g < N) hout[g * DF + col] = fmaxf(acc[r], 0.0f);
  }
}

// ----------------------------------------- attention scores (WMMA) + exp/denom
__global__ __launch_bounds__(256) void attn_kernel(
    const float* __restrict__ h,
    const float* __restrict__ Wa, const float* __restrict__ ba,
    const float* __restrict__ Wb, const float* __restrict__ bb,
    const float* __restrict__ Wc, const float* __restrict__ bc,
    const int* __restrict__ gid,
    float* __restrict__ evals, float* __restrict__ denom, int N) {
  __shared__ float hT[16 * PADR];
  __shared__ float rowacc[16];
  const int tid = threadIdx.x;
  const int wave = tid >> 5, lane = tid & 31;
  const int m = lane & 15, hf = lane >> 4;
  const long row0 = (long)blockIdx.x * 16;

  for (int idx = tid; idx < 512; idx += 256) {
    int r = idx >> 5, c4 = (idx & 31) << 2;
    long g = row0 + r;
    if (g >= N) g = N - 1;
    const float4 hv = *(const float4*)(h + g * DF + c4);
    float* hp = &hT[r * PADR + c4];
    hp[0] = hv.x; hp[1] = hv.y; hp[2] = hv.z; hp[3] = hv.w;
  }
  if (tid < 16) rowacc[tid] = 0.0f;
  __syncthreads();

#pragma unroll
  for (int dt = 0; dt < 2; ++dt) {
    const int t = (wave << 1) + dt;   // 16 column tiles of AH=256
    const int col = (t << 4) + m;
    v8f ca, cb;
#pragma unroll
    for (int r = 0; r < 8; ++r) { ca[r] = ba[col]; cb[r] = bb[col]; }
#pragma unroll
    for (int k0 = 0; k0 < DF; k0 += 4) {
      const int kb = k0 + (hf << 1);
      v2f a, w1, w2;
      a.x = hT[m * PADR + kb];
      a.y = hT[m * PADR + kb + 1];
      w1.x = Wa[kb * AHF + col];
      w1.y = Wa[(kb + 1) * AHF + col];
      ca = wmma_f32(a, w1, ca);
      w2.x = Wb[kb * AHF + col];
      w2.y = Wb[(kb + 1) * AHF + col];
      cb = wmma_f32(a, w2, cb);
    }
    const float wc = Wc[col];
#pragma unroll
    for (int r = 0; r < 8; ++r) {
      float av = tanhf(ca[r]);
      float bv = 1.0f / (1.0f + expf(-cb[r]));
      atomicAdd(&rowacc[r + (hf << 3)], av * bv * wc);
    }
  }
  __syncthreads();

  if (tid < 16) {
    long g = row0 + tid;
    if (g < N) {
      // e/s is invariant to the max-shift; |A|<=~16 so plain exp is fp32-safe.
      float e = expf(rowacc[tid] + bc[0]);
      evals[g] = e;
      atomicAdd(&denom[gid[g]], e);
    }
  }
}

// ------------------------------------------------ softmax-weighted pooling
__global__ __launch_bounds__(256) void pool_kernel(
    const float* __restrict__ h, const float* __restrict__ evals,
    const float* __restrict__ denom, const int* __restrict__ gid,
    float* __restrict__ out, int N) {
  __shared__ float pool[NGROUP * DF];  // 16x128
  const int tid = threadIdx.x;
  for (int idx = tid; idx < NGROUP * DF; idx += 256) pool[idx] = 0.0f;
  __syncthreads();

  const int sub = tid >> 5;              // 8 nodes in flight
  const int comp = (tid & 31) << 2;      // 4 floats per lane
  const long base = (long)blockIdx.x * 64;
  for (int it = 0; it < 8; ++it) {
    long i = base + (long)it * 8 + sub;
    if (i < N) {
      int g = gid[i];
      float w = evals[i] / denom[g];
      const float4 hv = *(const float4*)(h + i * DF + comp);
      float* p = &pool[g * DF + comp];
      atomicAdd(p + 0, w * hv.x);
      atomicAdd(p + 1, w * hv.y);
      atomicAdd(p + 2, w * hv.z);
      atomicAdd(p + 3, w * hv.w);
    }
  }
  __syncthreads();
  for (int idx = tid; idx < NGROUP * DF; idx += 256)
    atomicAdd(&out[idx], pool[idx]);
}

// ---------------------------------------------------------------- launcher
extern "C" void kernel_launch(void* const* d_in, const int* in_sizes, int n_in,
                              void* d_out, int out_size, void* d_ws, size_t ws_size,
                              hipStream_t stream) {
  const float* x       = (const float*)d_in[0];
  const float* W_self  = (const float*)d_in[1];
  const float* b_self  = (const float*)d_in[2];
  const float* W_neigh = (const float*)d_in[3];
  const float* b_neigh = (const float*)d_in[4];
  const float* Wa = (const float*)d_in[5];
  const float* ba = (const float*)d_in[6];
  const float* Wb = (const float*)d_in[7];
  const float* bb = (const float*)d_in[8];
  const float* Wc = (const float*)d_in[9];
  const float* bc = (const float*)d_in[10];
  const int* src = (const int*)d_in[11];
  const int* dst = (const int*)d_in[12];
  const int* gid = (const int*)d_in[13];
  const int N = in_sizes[0] / DF;
  const int E = in_sizes[11];
  (void)n_in; (void)ws_size;

  // workspace layout (floats): h | msg | deg | invd | evals | denom
  float* ws = (float*)d_ws;
  size_t off = 0;
  float* h    = ws + off; off += (size_t)N * DF;
  float* msg  = ws + off; off += (size_t)N * DF;
  float* deg  = ws + off; off += (size_t)N;
  float* invd = ws + off; off += (size_t)N;
  float* ev   = ws + off; off += (size_t)N;
  float* den  = ws + off; off += 64;

  const int nb_n   = (N + 255) / 256;
  const int nb_e   = (E + 255) / 256;
  const int nb_nd  = (int)(((long)N * DF + 255) / 256);
  const int nb_sc  = (int)(((long)E * 32 + 255) / 256);
  const int nb_row = (N + 15) / 16;

  fill_kernel<<<nb_n, 256, 0, stream>>>(deg, 0.0f, N);
  deg_kernel<<<nb_e, 256, 0, stream>>>(dst, deg, E);
  invdeg_kernel<<<nb_n, 256, 0, stream>>>(deg, invd, N);

  hipMemcpyAsync(h, x, (size_t)N * DF * sizeof(float),
                 hipMemcpyDeviceToDevice, stream);

  for (int l = 0; l < LAYERS; ++l) {
    fill_kernel<<<nb_nd, 256, 0, stream>>>(msg, 0.0f, (long)N * DF);
    scatter_kernel<<<nb_sc, 256, 0, stream>>>(h, src, dst, msg, E);
    layer_kernel<<<nb_row, 256, 0, stream>>>(
        h, msg, invd,
        W_self + (size_t)l * DF * DF, b_self + (size_t)l * DF,
        W_neigh + (size_t)l * DF * DF, b_neigh + (size_t)l * DF,
        h, N);  // in-place: each block reads only its own rows before writing
  }

  fill_kernel<<<1, 64, 0, stream>>>(den, 0.0f, 64);
  attn_kernel<<<nb_row, 256, 0, stream>>>(h, Wa, ba, Wb, bb, Wc, bc, gid, ev, den, N);

  fill_kernel<<<(NGROUP * DF + 255) / 256, 256, 0, stream>>>((float*)d_out, 0.0f,
                                                             NGROUP * DF);
  pool_kernel<<<(N + 63) / 64, 256, 0, stream>>>(h, ev, den, gid, (float*)d_out, N);
}